// HashGridEncoder_49641232007422
// MI455X (gfx1250) — compile-verified
//
#include <hip/hip_runtime.h>
#include <hip/hip_bf16.h>
#include <cstdint>

// ---------------------------------------------------------------------------
// HashGridEncoder for MI455X (gfx1250).
//
// Strategy: one level's feature table (32768 entries x 2 f32 = 256KB) fits in
// CDNA5's 320KB LDS. Each block = (one level, 8192 points). The table slice is
// bulk-copied global->LDS with the gfx1250 async-to-LDS path (ASYNCcnt), then
// all 8-corner hash gathers are served from LDS (ds_load_b64), not HBM/L2.
// No WMMA on purpose: the reduction is an 8-long data-dependent gather sum,
// not a matmul; the bottleneck is gather bandwidth/latency + VALU.
// ---------------------------------------------------------------------------

#define L_LEVELS       16
#define F_FEAT         2
#define T_SIZE         32768            // hash table entries per level
#define TPB            1024             // 32 wave32s -> 8 waves/SIMD32
#define PTS_PER_BLOCK  8192
#define SMEM_BYTES     (T_SIZE * F_FEAT * sizeof(float))   // 256 KB

// Pointee type required by the async-to-LDS b128 builtin: 4 x i32 vector.
typedef int v4i __attribute__((ext_vector_type(4)));
typedef __attribute__((address_space(1))) v4i* gv4i_p;   // global (prints as __device__)
typedef __attribute__((address_space(3))) v4i* lv4i_p;   // LDS    (prints as __shared__)

// 16B async copy global -> LDS (lowers to global_load_async_to_lds_b128,
// tracked by ASYNCcnt).
__device__ __forceinline__ void async_copy_b128(const void* gsrc, void* ldst) {
#if __has_builtin(__builtin_amdgcn_global_load_async_to_lds_b128)
    __builtin_amdgcn_global_load_async_to_lds_b128(
        (gv4i_p)(uintptr_t)gsrc,     // inttoptr: keeps the global VA, AS(1)
        (lv4i_p)ldst,                // addrspacecast generic->LDS: strips aperture
        /*offset=*/0, /*cpol=*/0);
#else
    unsigned lds_off = (unsigned)(uintptr_t)(lv4i_p)ldst;
    asm volatile("global_load_async_to_lds_b128 %0, %1, off"
                 :: "v"(lds_off), "v"(gsrc) : "memory");
#endif
}

__device__ __forceinline__ void wait_async_zero() {
#if __has_builtin(__builtin_amdgcn_s_wait_asynccnt)
    __builtin_amdgcn_s_wait_asynccnt(0);
#else
    asm volatile("s_wait_asynccnt 0x0" ::: "memory");
#endif
}

__global__ __launch_bounds__(TPB)
void HashGridEncoder_49641232007422_kernel(const float* __restrict__ x,
                                           const float* __restrict__ table,
                                           float* __restrict__ out,
                                           int n_points) {
    extern __shared__ float smem[];        // 256 KB: this level's table
    const int lvl = blockIdx.y;
    const int tid = threadIdx.x;

    // ---- Stage 1: async bulk copy of this level's 256KB table into LDS ----
    {
        const float4* gsrc = (const float4*)(table + (size_t)lvl * (T_SIZE * F_FEAT));
        float4*       ldst = (float4*)smem;
        const int nvec = (T_SIZE * F_FEAT) / 4;        // 16384 float4s
        // 16 async b128 instructions per thread; ASYNCcnt (6-bit) never
        // overflows a wave (16 < 63); HW stalls issue otherwise anyway.
        #pragma unroll 4
        for (int i = tid; i < nvec; i += TPB)
            async_copy_b128(gsrc + i, ldst + i);
    }
    wait_async_zero();       // per-wave: all my async loads landed in LDS
    __syncthreads();         // whole block: LDS table fully populated

    // scale_l = 16 * B^l - 1, B = (512/16)^(1/15) = 2^(1/3)
    const float scale = exp2f((float)lvl * (1.0f / 3.0f)) * 16.0f - 1.0f;

    const float2* lt = (const float2*)smem;
    const int n0 = blockIdx.x * PTS_PER_BLOCK;
    const int n1 = min(n0 + PTS_PER_BLOCK, n_points);

    for (int n = n0 + tid; n < n1; n += TPB) {
        const float px = x[3 * n + 0];
        const float py = x[3 * n + 1];
        const float pz = x[3 * n + 2];

        // xn = (x + 1) / 2 ; pos = xn * scale + 0.5
        const float posx = fmaf((px + 1.0f) * 0.5f, scale, 0.5f);
        const float posy = fmaf((py + 1.0f) * 0.5f, scale, 0.5f);
        const float posz = fmaf((pz + 1.0f) * 0.5f, scale, 0.5f);

        const float flx = floorf(posx), fly = floorf(posy), flz = floorf(posz);
        const float rx = posx - flx, ry = posy - fly, rz = posz - flz;
        const float ix = 1.0f - rx, iy = 1.0f - ry, iz = 1.0f - rz;

        const unsigned gx = (unsigned)(int)flx;
        const unsigned gy = (unsigned)(int)fly;
        const unsigned gz = (unsigned)(int)flz;

        // pre-hash the two values per dimension (PRIMES = {1, 2654435761, 805459861})
        const unsigned hx0 = gx,                hx1 = gx + 1u;
        const unsigned hy0 = gy * 2654435761u,  hy1 = (gy + 1u) * 2654435761u;
        const unsigned hz0 = gz * 805459861u,   hz1 = (gz + 1u) * 805459861u;

        float o0 = 0.0f, o1 = 0.0f;
        #pragma unroll
        for (int c = 0; c < 8; ++c) {
            const bool bx = (c >> 0) & 1;
            const bool by = (c >> 1) & 1;
            const bool bz = (c >> 2) & 1;
            const unsigned h = (bx ? hx1 : hx0) ^ (by ? hy1 : hy0) ^ (bz ? hz1 : hz0);
            const unsigned idx = h & (unsigned)(T_SIZE - 1);
            const float2 f = lt[idx];                         // ds_load_b64 from LDS
            const float w = (bx ? rx : ix) * (by ? ry : iy) * (bz ? rz : iz);
            o0 = fmaf(w, f.x, o0);
            o1 = fmaf(w, f.y, o1);
        }

        // out[n, L*F] layout: point n, level lvl -> 8B vector store
        float2* op = (float2*)(out + (size_t)n * (L_LEVELS * F_FEAT) + lvl * F_FEAT);
        *op = make_float2(o0, o1);
    }
}

extern "C" void kernel_launch(void* const* d_in, const int* in_sizes, int n_in,
                              void* d_out, int out_size, void* d_ws, size_t ws_size,
                              hipStream_t stream) {
    const float* x     = (const float*)d_in[0];   // (N, 3) f32
    const float* table = (const float*)d_in[1];   // (L, T, F) f32
    float*       out   = (float*)d_out;           // (N, L*F) f32

    const int n_points = in_sizes[0] / 3;

    // 256 KB dynamic LDS (> default cap) — raise the limit every call (idempotent).
    (void)hipFuncSetAttribute((const void*)HashGridEncoder_49641232007422_kernel,
                              hipFuncAttributeMaxDynamicSharedMemorySize,
                              (int)SMEM_BYTES);

    dim3 grid((n_points + PTS_PER_BLOCK - 1) / PTS_PER_BLOCK, L_LEVELS);
    dim3 block(TPB);
    HashGridEncoder_49641232007422_kernel<<<grid, block, SMEM_BYTES, stream>>>(
        x, table, out, n_points);
}